// NNConvAdj_49177375539506
// MI455X (gfx1250) — compile-verified
//
#include <hip/hip_runtime.h>
#include <cstddef>

typedef __attribute__((ext_vector_type(2))) float v2f;
typedef __attribute__((ext_vector_type(8))) float v8f;

#define B_    4
#define N_    256
#define FE_   8
#define CIN_  16
#define COUT_ 16
#define H_    64
#define JT    16            // j-tile per block
#define NWAVE 8             // 256 threads = 8 wave32
#define IPW   (N_ / NWAVE)  // i-values per wave

// ---------------------------------------------------------------------------
// out[b,j,d] = sum_c x[b,j,c] * ( (sum_i relu(e[b,i,j,:]W1+b1)) W2 + N*b2 )[c*16+d]
//            + sum_c x[b,j,c] * root[c,d]
//
// Inner loop accumulates 2*relu(x) = x + |x|  (bit-exact; |x| is a free VOP3
// source modifier, fadd needs no canonicalize). The 0.5 factor is folded into
// the linear cross-wave reduction.
// ---------------------------------------------------------------------------
__global__ __launch_bounds__(256, 1)
void nnconv_edge_fused(const float* __restrict__ x,     // [B,N,CIN]
                       const float* __restrict__ edge,  // [B,N,N,FE]
                       const float* __restrict__ W1,    // [FE,H]
                       const float* __restrict__ b1,    // [H]
                       const float* __restrict__ W2,    // [H,CIN*COUT]
                       const float* __restrict__ b2,    // [CIN*COUT]
                       const float* __restrict__ root,  // [CIN,COUT]
                       float* __restrict__ out)         // [B,N,COUT]
{
    __shared__ float lds_part[NWAVE][JT][H_];     // 32 KB: per-wave 2*Hsum partials
    __shared__ float lds_hsum[JT][H_];            //  4 KB: reduced Hsum
    __shared__ float lds_out [NWAVE][JT][COUT_];  //  8 KB: per-wave out partials

    const int tid  = threadIdx.x;
    const int wv   = tid >> 5;          // wave 0..7
    const int lane = tid & 31;
    const int lo   = (lane < 16) ? 1 : 0;
    const int n16  = lane & 15;

    const int bb = blockIdx.x / (N_ / JT);
    const int jt = blockIdx.x % (N_ / JT);
    const int j0 = jt * JT;

    // ---- preload W1 B-fragments (4x16 f32 B layout: b.x=B[k0][n], b.y=B[k0+1][n],
    //      k0 = kb*4 + (lane<16 ? 0 : 2)) and b1 bias values -------------------
    const int koffA = lo ? 0 : 2;
    v2f  w1frag[2][4];
    float bias1[4];
#pragma unroll
    for (int kb = 0; kb < 2; ++kb) {
        const int k0 = kb * 4 + koffA;
#pragma unroll
        for (int t = 0; t < 4; ++t) {
            const int col = t * 16 + n16;
            w1frag[kb][t].x = W1[(k0 + 0) * H_ + col];
            w1frag[kb][t].y = W1[(k0 + 1) * H_ + col];
        }
    }
#pragma unroll
    for (int t = 0; t < 4; ++t) bias1[t] = b1[t * 16 + n16];

    // ---- Stage 1: per-wave partial of 2*Hsum = sum_i (h + |h|), h = E@W1+b1 --
    v8f acc[4];
#pragma unroll
    for (int t = 0; t < 4; ++t)
#pragma unroll
        for (int v = 0; v < 8; ++v) acc[t][v] = 0.0f;

    // A fragment row m = lane%16 maps to node j0+m
    const float* ebase = edge + ((size_t)bb * N_ * N_ + (size_t)(j0 + n16)) * FE_;

#pragma unroll 2
    for (int ii = 0; ii < IPW; ++ii) {
        const int i = wv * IPW + ii;
        const float* ep = ebase + (size_t)i * (N_ * FE_);
        // A 16x4 f32 layout: a.x = A[m][k0], a.y = A[m][k0+1], k0 = lane<16?0:2
        v2f a0, a1;
        a0.x = ep[koffA + 0];
        a0.y = ep[koffA + 1];
        a1.x = ep[4 + koffA + 0];
        a1.y = ep[4 + koffA + 1];
#pragma unroll
        for (int t = 0; t < 4; ++t) {
            v8f c;
#pragma unroll
            for (int v = 0; v < 8; ++v) c[v] = bias1[t];
            c = __builtin_amdgcn_wmma_f32_16x16x4_f32(false, a0, false, w1frag[0][t],
                                                      (short)0, c, false, false);
            c = __builtin_amdgcn_wmma_f32_16x16x4_f32(false, a1, false, w1frag[1][t],
                                                      (short)0, c, false, false);
#pragma unroll
            for (int v = 0; v < 8; ++v)
                acc[t][v] += (c[v] + __builtin_fabsf(c[v]));   // 2*relu, exact
        }
    }

    // ---- cross-wave reduction (deterministic); fold the 0.5 factor here -----
    const int mrow = lo ? 0 : 8;  // C layout: row = v + (lane<16 ? 0 : 8)
#pragma unroll
    for (int t = 0; t < 4; ++t)
#pragma unroll
        for (int v = 0; v < 8; ++v)
            lds_part[wv][v + mrow][t * 16 + n16] = acc[t][v];
    __syncthreads();

#pragma unroll
    for (int e = 0; e < 4; ++e) {
        const int idx = tid + e * 256;     // 0..1023 over 16x64
        const int r   = idx >> 6;
        const int cc  = idx & 63;
        float s = 0.0f;
#pragma unroll
        for (int w = 0; w < NWAVE; ++w) s += lds_part[w][r][cc];
        lds_hsum[r][cc] = s * 0.5f;        // undo the 2x from (x + |x|)
    }
    __syncthreads();

    // ---- Stage 2: Wsum = Hsum @ W2 + N*b2, then out = x . (Wsum + root) -----
    // Wave wv owns output tiles t2 = 2wv, 2wv+1. Column q = t2*16 + n16 means
    // c = t2 and d = n16; C-layout row is j_local.
    float contrib[8];
#pragma unroll
    for (int v = 0; v < 8; ++v) contrib[v] = 0.0f;

#pragma unroll
    for (int tt = 0; tt < 2; ++tt) {
        const int t2 = wv * 2 + tt;                     // = c index
        v8f c;
        const float biasq = (float)N_ * b2[t2 * 16 + n16];
#pragma unroll
        for (int v = 0; v < 8; ++v) c[v] = biasq;
#pragma unroll
        for (int kb = 0; kb < 16; ++kb) {               // K = 64
            const int k0 = kb * 4 + koffA;
            v2f a, w2f;
            a.x = lds_hsum[n16][k0 + 0];                // A row m = lane%16
            a.y = lds_hsum[n16][k0 + 1];
            w2f.x = W2[(k0 + 0) * (CIN_ * COUT_) + t2 * 16 + n16];
            w2f.y = W2[(k0 + 1) * (CIN_ * COUT_) + t2 * 16 + n16];
            c = __builtin_amdgcn_wmma_f32_16x16x4_f32(false, a, false, w2f,
                                                      (short)0, c, false, false);
        }
        const float rt = root[t2 * COUT_ + n16];        // root[c, d]
#pragma unroll
        for (int v = 0; v < 8; ++v) {
            const int j = v + mrow;                     // j_local
            const float xv = x[((size_t)bb * N_ + j0 + j) * CIN_ + t2];
            contrib[v] += xv * (c[v] + rt);
        }
    }

#pragma unroll
    for (int v = 0; v < 8; ++v)
        lds_out[wv][v + mrow][n16] = contrib[v];
    __syncthreads();

    {
        const int j = tid >> 4;
        const int d = tid & 15;
        float s = 0.0f;
#pragma unroll
        for (int w = 0; w < NWAVE; ++w) s += lds_out[w][j][d];
        out[((size_t)bb * N_ + j0 + j) * COUT_ + d] = s;
    }
}

extern "C" void kernel_launch(void* const* d_in, const int* in_sizes, int n_in,
                              void* d_out, int out_size, void* d_ws, size_t ws_size,
                              hipStream_t stream) {
    const float* x    = (const float*)d_in[0];  // node_attr [B,N,CIN]
    const float* edge = (const float*)d_in[1];  // edge_adj  [B,N,N,FE]
    const float* W1   = (const float*)d_in[2];  // [FE,H]
    const float* b1   = (const float*)d_in[3];  // [H]
    const float* W2   = (const float*)d_in[4];  // [H,CIN*COUT]
    const float* b2   = (const float*)d_in[5];  // [CIN*COUT]
    const float* root = (const float*)d_in[6];  // [CIN,COUT]
    float* out = (float*)d_out;                 // [B,N,COUT]

    dim3 grid(B_ * (N_ / JT));   // 64 blocks
    dim3 block(256);             // 8 wave32
    nnconv_edge_fused<<<grid, block, 0, stream>>>(x, edge, W1, b1, W2, b2, root, out);

    (void)in_sizes; (void)n_in; (void)out_size; (void)d_ws; (void)ws_size;
}